// AutoEncoderGRU_22230750724521
// MI455X (gfx1250) — compile-verified
//
#include <hip/hip_runtime.h>
#include <stdint.h>

#define BB   4096            // batch
#define TT_  4096            // timesteps
#define HD   3               // hidden
#define BT   (BB * TT_)
#define BLK  128             // threads (=sequences) per fused block
#define NBLK (BB / BLK)      // 32 blocks
#define TILE 32              // timesteps per LDS tile
#define ROWP 33              // padded LDS row stride (odd -> bank-conflict-free)
#define TILEF (BLK * ROWP)   // floats per tile buffer

#if defined(__has_builtin)
#  if __has_builtin(__builtin_amdgcn_tensor_load_to_lds)
#    define HAVE_TDM 1
#  else
#    define HAVE_TDM 0
#  endif
#else
#  define HAVE_TDM 0
#endif

typedef unsigned int v4u  __attribute__((ext_vector_type(4)));
typedef int          v8i_ __attribute__((ext_vector_type(8)));
typedef int          v4i_ __attribute__((ext_vector_type(4)));
typedef float        v2f  __attribute__((ext_vector_type(2)));
typedef float        v8f  __attribute__((ext_vector_type(8)));

// ---------- fast math (short dependent chains) ----------
__device__ __forceinline__ float fast_exp(float x) {
    return __builtin_amdgcn_exp2f(x * 1.4426950408889634f);
}
__device__ __forceinline__ float sigf(float x) {
    return __builtin_amdgcn_rcpf(1.0f + fast_exp(-x));
}
__device__ __forceinline__ float tanh_fast(float x) {
    float e = fast_exp(2.0f * x);                               // exp(2x)
    return fmaf(-2.0f, __builtin_amdgcn_rcpf(e + 1.0f), 1.0f);  // 1 - 2/(e+1)
}

// ---------- CDNA5 async global->LDS (per-lane) ----------
__device__ __forceinline__ void async_ld_b32(uint32_t lds_off, const void* g) {
    uint64_t ga = (uint64_t)(uintptr_t)g;
    asm volatile("global_load_async_to_lds_b32 %0, %1, off"
                 :: "v"(lds_off), "v"(ga) : "memory");
}
__device__ __forceinline__ void wait_asynccnt0() {
    asm volatile("s_wait_asynccnt 0" ::: "memory");
}

#if HAVE_TDM
// ---------- CDNA5 Tensor Data Mover: one 128x32 f32 tile -> LDS ----------
// 2D descriptor; pad 1 DWORD every 32 DWORDs => LDS row stride = 33 floats (ROWP).
__device__ __forceinline__ void tdm_load_tile(uint64_t gaddr, uint32_t lds_addr) {
    v4u g0;
    g0[0] = 1u;                                                    // count=1 (valid), user mode
    g0[1] = lds_addr;                                              // LDS byte address
    g0[2] = (uint32_t)gaddr;                                       // global_addr[31:0]
    g0[3] = ((uint32_t)(gaddr >> 32) & 0x01FFFFFFu) | (2u << 30);  // addr[56:32] | type=2
    v8i_ g1;
    g1[0] = (int)((2u << 16)        // data_size = 4B
                | (1u << 20)        // pad_enable
                | (4u << 22)        // pad_interval code 4 = 32 DWORDs
                | (0u << 25));      // pad_amount  code 0 = 1 DWORD
    g1[1] = (int)((TT_ & 0xFFFFu) << 16);                 // tensor_dim0[15:0]
    g1[2] = (int)((TT_ >> 16) | ((BB & 0xFFFFu) << 16));  // tensor_dim0[31:16] | tensor_dim1[15:0]
    g1[3] = (int)((BB >> 16) | (TILE << 16));             // tensor_dim1[31:16] | tile_dim0=32
    g1[4] = (int)BLK;                                     // tile_dim1=128, tile_dim2=0
    g1[5] = (int)TT_;                                     // tensor_dim0_stride[31:0] = T
    g1[6] = 0;                                            // stride0 hi / stride1 lo
    g1[7] = 0;                                            // stride1 hi
    v4i_ gz = (v4i_)0;                                    // 2D tile: groups 2/3 unused
#if __clang_major__ >= 23
    v8i_ gz8 = (v8i_)0;
    __builtin_amdgcn_tensor_load_to_lds(g0, g1, gz, gz, gz8, 0);
#else
    __builtin_amdgcn_tensor_load_to_lds(g0, g1, gz, gz, 0);
#endif
}
#endif

// ---------- GRU cell, H=3, fully unrolled (PyTorch gate order r,z,n) ----------
__device__ __forceinline__ void gru_step(float x, float h[HD],
                                         const float wih[9], const float whh[27],
                                         const float bih[9], const float bhh[9]) {
    float r[3], z[3], n[3];
#pragma unroll
    for (int g = 0; g < 3; ++g) {
        float gi = fmaf(x, wih[g], bih[g]);
        float gh = bhh[g];
#pragma unroll
        for (int j = 0; j < 3; ++j) gh = fmaf(whh[g * 3 + j], h[j], gh);
        r[g] = sigf(gi + gh);
    }
#pragma unroll
    for (int g = 0; g < 3; ++g) {
        float gi = fmaf(x, wih[3 + g], bih[3 + g]);
        float gh = bhh[3 + g];
#pragma unroll
        for (int j = 0; j < 3; ++j) gh = fmaf(whh[(3 + g) * 3 + j], h[j], gh);
        z[g] = sigf(gi + gh);
    }
#pragma unroll
    for (int g = 0; g < 3; ++g) {
        float gi = fmaf(x, wih[6 + g], bih[6 + g]);
        float gh = bhh[6 + g];
#pragma unroll
        for (int j = 0; j < 3; ++j) gh = fmaf(whh[(6 + g) * 3 + j], h[j], gh);
        n[g] = tanh_fast(fmaf(r[g], gh, gi));
    }
#pragma unroll
    for (int g = 0; g < 3; ++g) h[g] = fmaf(z[g], h[g] - n[g], n[g]); // (1-z)n + z h
}

// ---------- fused encoder + decoder + partial loss ----------
__global__ void __launch_bounds__(BLK) fused_gru(
    const float* __restrict__ input, const int* __restrict__ seqlen,
    const float* __restrict__ eWih, const float* __restrict__ eWhh,
    const float* __restrict__ ebih, const float* __restrict__ ebhh,
    const float* __restrict__ dWih, const float* __restrict__ dWhh,
    const float* __restrict__ dbih, const float* __restrict__ dbhh,
    const float* __restrict__ linW, const float* __restrict__ linb,
    float* __restrict__ out_y, float* __restrict__ out_feat,
    float* __restrict__ ws) {
    __shared__ float smem[2 * TILEF];
    __shared__ int   sseq[BLK];

    const int tid = threadIdx.x;
    const int b0  = blockIdx.x * BLK;
    const int b   = b0 + tid;

    const uint32_t lds_base = (uint32_t)(uintptr_t)(&smem[0]); // low 32 bits = LDS offset
    const uint32_t seq_base = (uint32_t)(uintptr_t)(&sseq[0]);
    const uint64_t gbase    = (uint64_t)(uintptr_t)(input + (size_t)b0 * TT_);

#if !HAVE_TDM
    // fallback: per-lane async B32 copies; flat = i*BLK+tid -> row=flat>>5, col=flat&31,
    // consecutive lanes hit consecutive t of one row: fully coalesced 128B/wave.
    auto issue_tile = [&](int k, int bufsel) {
        const int t0 = k * TILE;
#pragma unroll 4
        for (int i = 0; i < TILE; ++i) {
            int flat = i * BLK + tid;
            int r = flat >> 5;
            int c = flat & 31;
            uint32_t loff = lds_base + (uint32_t)((bufsel * TILEF + r * ROWP + c) * 4);
            async_ld_b32(loff, input + (size_t)(b0 + r) * TT_ + (t0 + c));
        }
    };
#endif

    // --- stage seq_lengths slice through LDS with the per-lane async path ---
    async_ld_b32(seq_base + (uint32_t)tid * 4u, seqlen + b);

    // --- encoder weights into registers (uniform -> SGPRs) ---
    float wih[9], whh[27], bih[9], bhh[9];
#pragma unroll
    for (int i = 0; i < 9; ++i) { wih[i] = eWih[i]; bih[i] = ebih[i]; bhh[i] = ebhh[i]; }
#pragma unroll
    for (int i = 0; i < 27; ++i) whh[i] = eWhh[i];

    // --- prologue: tile 0 ---
#if HAVE_TDM
    if (tid < 32) {                        // one wave issues the DMA (EXEC-independent op)
        tdm_load_tile(gbase, lds_base);
        __builtin_amdgcn_s_wait_tensorcnt(0);
    }
#else
    issue_tile(0, 0);
#endif
    wait_asynccnt0();                      // seq_lengths (and fallback tile) landed
    __syncthreads();

    const int myseq = sseq[tid];

    float h[HD] = {0.f, 0.f, 0.f};
    const int NT = TT_ / TILE; // 128 tiles
    for (int k = 0; k < NT; ++k) {
        int cur = k & 1;
        if (k + 1 < NT) {                  // prefetch next tile (double buffer)
#if HAVE_TDM
            if (tid < 32)
                tdm_load_tile(gbase + (uint64_t)(k + 1) * (TILE * 4),
                              lds_base + (uint32_t)((cur ^ 1) * TILEF * 4));
#else
            issue_tile(k + 1, cur ^ 1);
#endif
        }
        int t0  = k * TILE;
        int rem = myseq - t0;
        int jmax = rem >= TILE ? TILE : (rem > 0 ? rem : 0); // pack_padded: freeze at seq end
        const float* row = &smem[cur * TILEF + tid * ROWP];
        for (int j = 0; j < jmax; ++j)
            gru_step(row[j], h, wih, whh, bih, bhh);
#if HAVE_TDM
        if (tid < 32) __builtin_amdgcn_s_wait_tensorcnt(0); // next tile committed to LDS
#else
        wait_asynccnt0();
#endif
        __syncthreads();                   // ...visible to every wave
    }

    // features = sigmoid(h_final)
    float f0 = sigf(h[0]), f1 = sigf(h[1]), f2 = sigf(h[2]);
    out_feat[b * 3 + 0] = f0;
    out_feat[b * 3 + 1] = f1;
    out_feat[b * 3 + 2] = f2;

    // --- decoder weights ---
#pragma unroll
    for (int i = 0; i < 9; ++i) { wih[i] = dWih[i]; bih[i] = dbih[i]; bhh[i] = dbhh[i]; }
#pragma unroll
    for (int i = 0; i < 27; ++i) whh[i] = dWhh[i];
    float lw0 = linW[0], lw1 = linW[1], lw2 = linW[2], lb = linb[0];

    float hd[HD] = {f0, f1, f2};
    float x = 0.0f;
    float sse = 0.0f;
    const float* inrow = input + (size_t)b * TT_; // L2-resident reread for the loss
    float*       orow  = out_y + (size_t)b * TT_;
    for (int t = 0; t < TT_; ++t) {
        gru_step(x, hd, wih, whh, bih, bhh);
        float y = fmaf(lw2, hd[2], fmaf(lw1, hd[1], fmaf(lw0, hd[0], lb)));
        orow[t] = y;
        if (t < myseq) { float d = inrow[t] - y; sse = fmaf(d, d, sse); }
        x = y;
    }

    // --- block reduction of (sse, valid-count) via FP32 WMMA ones-vector trick ---
    // D = ones(16x4) x B(4x16) + C computes column sums: two accumulator-chained
    // v_wmma_f32_16x16x4_f32 reduce 128 values -> 16 exact FP32 partials.
    __syncthreads();
    smem[tid]       = sse;
    smem[BLK + tid] = (float)myseq;   // sum(seq_len) < 2^24: exact in f32
    __syncthreads();
    if (tid < 32) {                   // wave 0 only; EXEC all ones as WMMA requires
        v2f onesA; onesA[0] = 1.0f; onesA[1] = 1.0f;   // A(16x4) all-ones
        v8f zc = {};
        v2f b0v, b1v, b2v, b3v;
        b0v[0] = smem[tid];             b0v[1] = smem[tid + 32];        // sse[0..63]
        b1v[0] = smem[tid + 64];        b1v[1] = smem[tid + 96];        // sse[64..127]
        b2v[0] = smem[BLK + tid];       b2v[1] = smem[BLK + tid + 32];  // cnt[0..63]
        b3v[0] = smem[BLK + tid + 64];  b3v[1] = smem[BLK + tid + 96];  // cnt[64..127]
        v8f c1 = __builtin_amdgcn_wmma_f32_16x16x4_f32(false, onesA, false, b0v,
                                                       (short)0, zc, false, false);
        c1     = __builtin_amdgcn_wmma_f32_16x16x4_f32(false, onesA, false, b1v,
                                                       (short)0, c1, false, false);
        v8f c2 = __builtin_amdgcn_wmma_f32_16x16x4_f32(false, onesA, false, b2v,
                                                       (short)0, zc, false, false);
        c2     = __builtin_amdgcn_wmma_f32_16x16x4_f32(false, onesA, false, b3v,
                                                       (short)0, c2, false, false);
        // c[0]: col-sum for column (lane%16); columns 0..15 partition all 128 values
        smem[256 + tid] = c1[0];
        smem[320 + tid] = c2[0];
    }
    __syncthreads();
    if (tid == 0) {
        float s = 0.f, c = 0.f;
        for (int i = 0; i < 16; ++i) { s += smem[256 + i]; c += smem[320 + i]; }
        ws[blockIdx.x]        = s;
        ws[NBLK + blockIdx.x] = c;
    }
}

// ---------- fixed-order reduction of 32 partials -> loss ----------
__global__ void finalize_loss(const float* __restrict__ ws, float* __restrict__ loss) {
    if (threadIdx.x == 0 && blockIdx.x == 0) {
        float s = 0.f, c = 0.f;
        for (int i = 0; i < NBLK; ++i) { s += ws[i]; c += ws[NBLK + i]; }
        loss[0] = s / c;
    }
}

// ---------- coalesced echo of input into d_out ----------
__global__ void copy_input(const float4* __restrict__ in, float* __restrict__ out, int n4) {
    int i = blockIdx.x * blockDim.x + threadIdx.x;
    if (i < n4) {
        float4 v = in[i];
        out[4 * i + 0] = v.x;
        out[4 * i + 1] = v.y;  // out is only 4B-aligned (offset 1 float) -> scalar stores
        out[4 * i + 2] = v.z;
        out[4 * i + 3] = v.w;
    }
}

extern "C" void kernel_launch(void* const* d_in, const int* in_sizes, int n_in,
                              void* d_out, int out_size, void* d_ws, size_t ws_size,
                              hipStream_t stream) {
    (void)in_sizes; (void)n_in; (void)out_size; (void)ws_size;
    const float* input = (const float*)d_in[0];
    const int*   seql  = (const int*)d_in[1];
    const float* eWih  = (const float*)d_in[2];
    const float* eWhh  = (const float*)d_in[3];
    const float* ebih  = (const float*)d_in[4];
    const float* ebhh  = (const float*)d_in[5];
    const float* dWih  = (const float*)d_in[6];
    const float* dWhh  = (const float*)d_in[7];
    const float* dbih  = (const float*)d_in[8];
    const float* dbhh  = (const float*)d_in[9];
    const float* linW  = (const float*)d_in[10];
    const float* linb  = (const float*)d_in[11];
    float* out = (float*)d_out;
    float* ws  = (float*)d_ws;

    // d_out layout: [loss | input (BT) | output (BT) | features (B*H)]
    float* out_input  = out + 1;
    float* out_output = out + 1 + (size_t)BT;
    float* out_feat   = out + 1 + 2 * (size_t)BT;

    copy_input<<<(BT / 4 + 255) / 256, 256, 0, stream>>>((const float4*)input, out_input, BT / 4);
    fused_gru<<<NBLK, BLK, 0, stream>>>(input, seql, eWih, eWhh, ebih, ebhh,
                                        dWih, dWhh, dbih, dbhh, linW, linb,
                                        out_output, out_feat, ws);
    finalize_loss<<<1, 32, 0, stream>>>(ws, out);
}